// SpecialSpmm_13374528160096
// MI455X (gfx1250) — compile-verified
//
#include <hip/hip_runtime.h>
#include <stdint.h>

#define EPB   1024   // edges per LDS tile
#define WAVES 8
#define TPB   256
#define BATCH 8      // edges in flight per wave (hand-unrolled)

typedef __attribute__((ext_vector_type(4))) unsigned int v4u;
typedef __attribute__((ext_vector_type(8))) int          v8i;
typedef __attribute__((ext_vector_type(4))) int          v4i;

// Generic pointers to __shared__ carry the LDS byte offset in their low 32 bits
// (shared aperture base has zero low bits), which is what the D# lds_addr wants.
__device__ __forceinline__ uint32_t lds_offset(const void* p) {
    return (uint32_t)(uintptr_t)p;
}

// Issue one TDM tensor_load_to_lds for a 2-D tile of 4-byte elements.
//   D# group0: count=1 | lds_addr | global_addr[56:0] | type=2
//   D# group1: data_size=4B, tensor_dim0/1, tile_dim0/1, tensor_dim0_stride
__device__ __forceinline__ void tdm_load_2d(uint32_t lds_off, const void* gptr,
                                            uint32_t tensor_d0, uint32_t tensor_d1,
                                            uint64_t d0_stride,
                                            uint32_t tile_d0, uint32_t tile_d1) {
    uint64_t ga = (uint64_t)(uintptr_t)gptr;
    v4u g0;
    g0[0] = 1u;                                                  // count=1, user mode
    g0[1] = lds_off;                                             // lds_addr (bytes)
    g0[2] = (uint32_t)ga;                                        // global_addr[31:0]
    g0[3] = (uint32_t)((ga >> 32) & 0x01FFFFFFu) | (2u << 30);   // addr[56:32] | type=2
    v8i g1;
    g1[0] = (int)(2u << 16);                                     // data_size = 4 bytes
    g1[1] = (int)(tensor_d0 << 16);                              // tensor_dim0[15:0] -> bits 63:48
    g1[2] = (int)((tensor_d0 >> 16) | (tensor_d1 << 16));        // dim0[31:16], dim1[15:0]
    g1[3] = (int)((tensor_d1 >> 16) | (tile_d0 << 16));          // dim1[31:16], tile_dim0
    g1[4] = (int)(tile_d1 & 0xFFFFu);                            // tile_dim1 (tile_dim2=0)
    g1[5] = (int)(uint32_t)d0_stride;                            // tensor_dim0_stride[31:0]
    g1[6] = (int)((uint32_t)(d0_stride >> 32) & 0xFFFFu);        // stride[47:32]
    g1[7] = 0;
    v4i gz = (v4i){0, 0, 0, 0};
#if __clang_major__ >= 23
    v8i gz8 = (v8i){0, 0, 0, 0, 0, 0, 0, 0};
    __builtin_amdgcn_tensor_load_to_lds(g0, g1, gz, gz, gz8, 0);
#else
    __builtin_amdgcn_tensor_load_to_lds(g0, g1, gz, gz, 0);
#endif
}

// Load phase: wave-uniform row/col via readfirstlane (-> SADDR addressing),
// one coalesced 512B gather of b's row per edge.
#define EDGE_LOAD(u)                                                              \
    const int    row##u = __builtin_amdgcn_readfirstlane(rows[ebase + (u)*WAVES]);\
    const int    col##u = __builtin_amdgcn_readfirstlane(cols[ebase + (u)*WAVES]);\
    const float  v##u   = vals[ebase + (u)*WAVES];                                \
    const float4 bb##u  = b4[(size_t)col##u * 32 + lane];

// Update phase: 4 no-return fp32 atomics per lane (resolve in L2).
#define EDGE_STORE(u) {                                                           \
    float* op = out + (size_t)row##u * 128 + lane * 4;                            \
    atomicAdd(op + 0, v##u * bb##u.x);                                            \
    atomicAdd(op + 1, v##u * bb##u.y);                                            \
    atomicAdd(op + 2, v##u * bb##u.z);                                            \
    atomicAdd(op + 3, v##u * bb##u.w); }

__global__ void __launch_bounds__(TPB)
spmm_coo_kernel(const float* __restrict__ values,
                const float* __restrict__ b,
                const int*   __restrict__ indices,   // [2][E]: rows then cols
                float*       __restrict__ out,
                int E, int ntiles)
{
    __shared__ int   s_idx[2][2][EPB];   // [buf][row/col][edge]
    __shared__ float s_val[2][EPB];      // [buf][edge]

    const int  tid    = threadIdx.x;
    const int  wid    = tid >> 5;
    const int  lane   = tid & 31;
    const bool issuer = (tid < 32);      // wave 0 owns the TDM queue

    const float4* __restrict__ b4 = (const float4*)b;

    int t = blockIdx.x;
    if (t >= ntiles) return;

    // Prologue: DMA tile t into buffer 0 (indices as a 2xEPB tile, values as 1xEPB).
    if (issuer) {
        const int      base = t * EPB;
        const uint32_t rem  = (uint32_t)(E - base);   // OOB reads clamp to zero
        tdm_load_2d(lds_offset(&s_idx[0][0][0]), indices + base, rem, 2u,
                    (uint64_t)(uint32_t)E, EPB, 2u);
        tdm_load_2d(lds_offset(&s_val[0][0]),    values  + base, rem, 1u,
                    (uint64_t)EPB, EPB, 1u);
        __builtin_amdgcn_s_wait_tensorcnt(0);
    }
    __syncthreads();

    int buf = 0;
    for (; t < ntiles; t += gridDim.x) {
        // Kick off the DMA for the next tile before touching this one.
        const int tn = t + gridDim.x;
        if (tn < ntiles && issuer) {
            const int      base = tn * EPB;
            const uint32_t rem  = (uint32_t)(E - base);
            tdm_load_2d(lds_offset(&s_idx[buf ^ 1][0][0]), indices + base, rem, 2u,
                        (uint64_t)(uint32_t)E, EPB, 2u);
            tdm_load_2d(lds_offset(&s_val[buf ^ 1][0]),    values  + base, rem, 1u,
                        (uint64_t)EPB, EPB, 1u);
        }

        const int* __restrict__   rows = &s_idx[buf][0][0];
        const int* __restrict__   cols = &s_idx[buf][1][0];
        const float* __restrict__ vals = &s_val[buf][0];

        const int remain = E - t * EPB;
        if (remain >= EPB) {
            // Full tile: 128 edges per wave, hand-unrolled in batches of 8 so
            // 8 independent b-row gathers are in flight before any atomics
            // (the compiler's unroller refuses loops around convergent
            //  readfirstlane; straight-line code has no such restriction).
            for (int i = 0; i < EPB / (WAVES * BATCH); ++i) {   // 16 iterations
                const int ebase = wid + i * (WAVES * BATCH);
                EDGE_LOAD(0) EDGE_LOAD(1) EDGE_LOAD(2) EDGE_LOAD(3)
                EDGE_LOAD(4) EDGE_LOAD(5) EDGE_LOAD(6) EDGE_LOAD(7)
                EDGE_STORE(0) EDGE_STORE(1) EDGE_STORE(2) EDGE_STORE(3)
                EDGE_STORE(4) EDGE_STORE(5) EDGE_STORE(6) EDGE_STORE(7)
            }
        } else {
            // Last (partial) tile only.
            for (int e = wid; e < remain; e += WAVES) {
                const int   row = __builtin_amdgcn_readfirstlane(rows[e]);
                const int   col = __builtin_amdgcn_readfirstlane(cols[e]);
                const float v   = vals[e];
                const float4 bb = b4[(size_t)col * 32 + lane];
                float* op = out + (size_t)row * 128 + lane * 4;
                atomicAdd(op + 0, v * bb.x);
                atomicAdd(op + 1, v * bb.y);
                atomicAdd(op + 2, v * bb.z);
                atomicAdd(op + 3, v * bb.w);
            }
        }

        if (issuer) __builtin_amdgcn_s_wait_tensorcnt(0);
        __syncthreads();     // next-tile data landed AND everyone done with this buffer
        buf ^= 1;
    }
}

__global__ void __launch_bounds__(TPB) spmm_zero_kernel(float4* __restrict__ out, long n4) {
    long i = (long)blockIdx.x * TPB + threadIdx.x;
    const long stride = (long)gridDim.x * TPB;
    const float4 z = make_float4(0.f, 0.f, 0.f, 0.f);
    for (; i < n4; i += stride) out[i] = z;
}

extern "C" void kernel_launch(void* const* d_in, const int* in_sizes, int n_in,
                              void* d_out, int out_size, void* d_ws, size_t ws_size,
                              hipStream_t stream) {
    const float* values  = (const float*)d_in[0];
    const float* b       = (const float*)d_in[1];
    const int*   indices = (const int*)d_in[2];
    float*       out     = (float*)d_out;

    const int E = in_sizes[0];            // 1.6M edges

    // out is poisoned by the harness; zero it before the atomic accumulation.
    const long n4 = (long)out_size / 4;
    spmm_zero_kernel<<<dim3(1024), dim3(TPB), 0, stream>>>((float4*)d_out, n4);

    const int ntiles = (E + EPB - 1) / EPB;
    const int grid   = ntiles < 768 ? ntiles : 768;
    spmm_coo_kernel<<<dim3(grid), dim3(TPB), 0, stream>>>(values, b, indices, out, E, ntiles);
}